// SideChainBuilder_40819369181403
// MI455X (gfx1250) — compile-verified
//
#include <hip/hip_runtime.h>
#include <hip/hip_bf16.h>

// SideChainBuilder for MI455X (gfx1250).
// Memory-bound NeRF sidechain extension: no matmul structure -> WMMA not
// applicable. CDNA5-specific paths used instead:
//   * global_load_async_to_lds_b32 (ASYNCcnt) to stage per-AA tables into LDS
//   * s_wait_asynccnt split counter
//   * wave32 bank-conflict-free LDS layout used as runtime-indexable
//     per-thread atom storage (parents are data-dependent indices).

#define NB     64
#define NR     8192
#define NRES   (NB * NR)      // 524288 = 2048 * 256 exactly
#define NATOM  10
#define MAXA   14
#define NAA    20
#define NCHI   4
#define BLOCK  256
#define SLAB   (BLOCK * 3)    // dword stride between atom slabs (768 % 64 == 0)
#define EPS    1e-6f

// Async copy of `ndw` dwords from global to LDS, striped over the block.
// ISA: GLOBAL_LOAD_ASYNC_TO_LDS_B32 (VGLOBAL op 96), tracked by ASYNCcnt.
__device__ __forceinline__ void async_copy_dwords(void* lds_dst, const void* gsrc,
                                                  int ndw, int tid) {
  // Low 32 bits of a generic LDS address are the LDS byte offset (aperture in
  // the high bits), so truncation yields the DS address the instruction wants.
  uint32_t lds_base = (uint32_t)(uintptr_t)lds_dst;
  const uint32_t* g = (const uint32_t*)gsrc;
  for (int i = tid; i < ndw; i += BLOCK) {
    uint32_t ldsa = lds_base + (uint32_t)(i * 4);
    uint64_t ga   = (uint64_t)(uintptr_t)(g + i);
    asm volatile("global_load_async_to_lds_b32 %0, %1, off"
                 :: "v"(ldsa), "v"(ga) : "memory");
  }
}

__device__ __forceinline__ void wait_async0() {
#if __has_builtin(__builtin_amdgcn_s_wait_asynccnt)
  __builtin_amdgcn_s_wait_asynccnt(0);
#else
  asm volatile("s_wait_asynccnt 0" ::: "memory");
#endif
}

extern "C" __global__ void __launch_bounds__(BLOCK)
sidechain_builder_kernel(const float* __restrict__ X,
                         const int*   __restrict__ C,
                         const int*   __restrict__ S,
                         const float* __restrict__ chi,
                         const float* __restrict__ Zt,     // [3][10][20] f32
                         const int*   __restrict__ Pt,     // [3][10][20] i32
                         const int*   __restrict__ chiIx,  // [4][20] i32
                         const int*   __restrict__ apa,    // [20] i32
                         float* __restrict__ outX,         // [NRES][14][3]
                         float* __restrict__ outM)         // [NRES][14]
{
  __shared__ float sZ[3 * NATOM * NAA];     // 600 dwords
  __shared__ int   sP[3 * NATOM * NAA];     // 600 dwords
  __shared__ int   sCI[NCHI * NAA];         // 80 dwords
  __shared__ int   sA[NAA];                 // 20 dwords
  // [atom][tid][comp]: slab stride 768 dwords (multiple of 64 banks), lane
  // term 3*t distinct mod 64 -> conflict-free for divergent atom indices.
  __shared__ float sAtoms[MAXA * SLAB];     // 43008 B

  const int tid = threadIdx.x;
  const int idx = blockIdx.x * BLOCK + tid;   // grid covers NRES exactly

  // Kick off table staging (ASYNCcnt), overlap with per-residue loads below.
  async_copy_dwords(sZ,  Zt,    3 * NATOM * NAA, tid);
  async_copy_dwords(sP,  Pt,    3 * NATOM * NAA, tid);
  async_copy_dwords(sCI, chiIx, NCHI * NAA,      tid);
  async_copy_dwords(sA,  apa,   NAA,             tid);

  const int s     = S[idx];
  const int cflag = C[idx];
  const float4* xv = (const float4*)X + (size_t)idx * 3;   // 48 B/residue, 16B aligned
  const float4 x0 = xv[0];
  const float4 x1 = xv[1];
  const float4 x2 = xv[2];
  const float4 ch = ((const float4*)chi)[idx];

  wait_async0();
  __syncthreads();

  float* my = &sAtoms[tid * 3];   // atom a, comp c -> my[a*SLAB + c]
  my[0 * SLAB + 0] = x0.x; my[0 * SLAB + 1] = x0.y; my[0 * SLAB + 2] = x0.z;
  my[1 * SLAB + 0] = x0.w; my[1 * SLAB + 1] = x1.x; my[1 * SLAB + 2] = x1.y;
  my[2 * SLAB + 0] = x1.z; my[2 * SLAB + 1] = x1.w; my[2 * SLAB + 2] = x2.x;
  my[3 * SLAB + 0] = x2.y; my[3 * SLAB + 1] = x2.z; my[3 * SLAB + 2] = x2.w;

  // Dihedrals from table, then chi overrides at chi_ix slots (values in [4,10)).
  float D[NATOM];
#pragma unroll
  for (int i = 0; i < NATOM; ++i) D[i] = sZ[2 * (NATOM * NAA) + i * NAA + s];
  const float chv[4] = {ch.x, ch.y, ch.z, ch.w};
#pragma unroll
  for (int k = 0; k < NCHI; ++k) {
    const int j = sCI[k * NAA + s];
#pragma unroll
    for (int t = 0; t < NATOM; ++t)
      if (j == t) D[t] = chv[k];
  }

  // Sequential NeRF chain; each lane touches only its own LDS slab, so no
  // barriers are needed inside the loop (per-wave DS ordering suffices).
#pragma unroll
  for (int i = 0; i < NATOM; ++i) {
    const int p0 = sP[0 * (NATOM * NAA) + i * NAA + s];
    const int p1 = sP[1 * (NATOM * NAA) + i * NAA + s];
    const int p2 = sP[2 * (NATOM * NAA) + i * NAA + s];

    const float X1x = my[p0 * SLAB + 0], X1y = my[p0 * SLAB + 1], X1z = my[p0 * SLAB + 2];
    const float X2x = my[p1 * SLAB + 0], X2y = my[p1 * SLAB + 1], X2z = my[p1 * SLAB + 2];
    const float X3x = my[p2 * SLAB + 0], X3y = my[p2 * SLAB + 1], X3z = my[p2 * SLAB + 2];

    const float Bl = sZ[0 * (NATOM * NAA) + i * NAA + s];
    const float Aa = sZ[1 * (NATOM * NAA) + i * NAA + s];
    const float Dd = D[i];

    // bc = normalize(X3 - X2)
    float bcx = X3x - X2x, bcy = X3y - X2y, bcz = X3z - X2z;
    const float rb = rsqrtf(bcx * bcx + bcy * bcy + bcz * bcz + EPS);
    bcx *= rb; bcy *= rb; bcz *= rb;

    // n = normalize(cross(X2 - X1, bc))
    const float ux = X2x - X1x, uy = X2y - X1y, uz = X2z - X1z;
    float nx = uy * bcz - uz * bcy;
    float ny = uz * bcx - ux * bcz;
    float nz = ux * bcy - uy * bcx;
    const float rn = rsqrtf(nx * nx + ny * ny + nz * nz + EPS);
    nx *= rn; ny *= rn; nz *= rn;

    // m = cross(n, bc)
    const float mx = ny * bcz - nz * bcy;
    const float my_ = nz * bcx - nx * bcz;
    const float mz = nx * bcy - ny * bcx;

    const float sA_ = __sinf(Aa);
    const float d0  = -__cosf(Aa);
    const float d1  = sA_ * __cosf(Dd);
    const float d2  = sA_ * __sinf(Dd);

    const float X4x = X3x + Bl * (d0 * bcx + d1 * mx  + d2 * nx);
    const float X4y = X3y + Bl * (d0 * bcy + d1 * my_ + d2 * ny);
    const float X4z = X3z + Bl * (d0 * bcz + d1 * mz  + d2 * nz);

    my[(4 + i) * SLAB + 0] = X4x;
    my[(4 + i) * SLAB + 1] = X4y;
    my[(4 + i) * SLAB + 2] = X4z;
  }

  // Masked writeback: mask[a] = (a < atoms_per_aa[s] * (C>0))
  const int apr = (cflag > 0) ? sA[s] : 0;
  float* o  = outX + (size_t)idx * (MAXA * 3);
  float* om = outM + (size_t)idx * MAXA;
#pragma unroll
  for (int a = 0; a < MAXA; ++a) {
    const float mk = (a < apr) ? 1.0f : 0.0f;
    o[a * 3 + 0] = my[a * SLAB + 0] * mk;
    o[a * 3 + 1] = my[a * SLAB + 1] * mk;
    o[a * 3 + 2] = my[a * SLAB + 2] * mk;
    om[a] = mk;
  }
}

extern "C" void kernel_launch(void* const* d_in, const int* in_sizes, int n_in,
                              void* d_out, int out_size, void* d_ws, size_t ws_size,
                              hipStream_t stream) {
  const float* X     = (const float*)d_in[0];   // (64,8192,4,3) f32
  const int*   C     = (const int*)  d_in[1];   // (64,8192) i32
  const int*   S     = (const int*)  d_in[2];   // (64,8192) i32
  const float* chi   = (const float*)d_in[3];   // (64,8192,4) f32
  const float* Zt    = (const float*)d_in[4];   // (3,10,20) f32
  const int*   Pt    = (const int*)  d_in[5];   // (3,10,20) i32
  const int*   cix   = (const int*)  d_in[6];   // (4,20) i32
  const int*   apa   = (const int*)  d_in[7];   // (20,) i32

  float* outX = (float*)d_out;                           // NRES*14*3
  float* outM = outX + (size_t)NRES * (MAXA * 3);        // NRES*14

  sidechain_builder_kernel<<<NRES / BLOCK, BLOCK, 0, stream>>>(
      X, C, S, chi, Zt, Pt, cix, apa, outX, outM);
}